// RefinementLayer2_48945447306024
// MI455X (gfx1250) — compile-verified
//
#include <hip/hip_runtime.h>

// Problem constants (from reference): B=2, S=384, C=4, H=128, D=768
#define BB 2
#define SS 384
#define CC 4
#define HH 128
#define DD 768

typedef float v2f __attribute__((ext_vector_type(2)));
typedef float v8f __attribute__((ext_vector_type(8)));

// Fast tanh via hardware transcendentals: tanh(x) = 1 - 2/(exp2(2x*log2e)+1).
// Saturates correctly at +-1 for large |x| (exp->inf => rcp->0, exp->0 => rcp->1).
__device__ __forceinline__ float tanhf_fast(float x) {
  float e = __builtin_amdgcn_exp2f(x * 2.8853900817779268f); // 2*log2(e)
  float r = __builtin_amdgcn_rcpf(e + 1.0f);
  return 1.0f - 2.0f * r;
}

// ---------------- Epilogue modes (compile-time) ----------------
#define MODE_TANH 1  // store tanh(v) to O
#define MODE_BIAS 4  // v += bias[col]
#define MODE_ADDV 8  // v += addvec[(row/C)*H + col]   (fuses t_ref + b_mid into pref)

// Store one 16x16 f32 C/D fragment. ISA: VGPR i -> row m0+i (lanes 0-15) or m0+i+8 (lanes 16-31),
// col = n0 + (lane&15).
template <int MODE, bool HAS_RAW>
__device__ __forceinline__ void store_tile(v8f c, float* O, int ldo, float* Oraw,
                                           const float* bias, const float* addvec,
                                           int m0, int n0, int lane) {
  int col   = n0 + (lane & 15);
  int rbase = m0 + ((lane >> 4) << 3);
#pragma unroll
  for (int i = 0; i < 8; ++i) {
    int   row = rbase + i;
    float v   = c[i];
    if constexpr (MODE & MODE_BIAS) v += bias[col];
    if constexpr (MODE & MODE_ADDV) v += addvec[(long)(row >> 2) * HH + col]; // row/C, C==4
    if constexpr (HAS_RAW) Oraw[(long)row * ldo + col] = v;
    O[(long)row * ldo + col] = (MODE & MODE_TANH) ? tanhf_fast(v) : v;
  }
}

// ---------------- Generic batched LDS-staged WMMA GEMM (full f32) ----------------
// Block = 4 wave32 waves; block tile 64x64, each wave owns a 32x32 sub-tile
// (2x2 of 16x16 f32 accumulators), K staged in chunks of 16 through LDS.
//
// LDS layout (leading dim 20 floats):
//   As[m][k]  m=0..63, k=0..15   (row-major, matches global A chunk rows)
//   Bs[n][k]  n=0..63, k=0..15   (B chunk stored TRANSPOSED so B-fragment reads
//                                 B[k][n],B[k+1][n] become one aligned 64-bit LDS read)
// ld=20: (20*m + k) mod 64 distinct over a 16-lane group (gcd(20,64)=4, period 16)
// -> conflict-free ds_load_b64 fragment reads; 20*4 bytes = 80 ≡ 0 (mod 16) keeps
// the float4 staging stores 16B-aligned.
//
// Batch z decodes as (z/C, z%C) with separate outer/inner offsets (C==4).
template <int MODE, bool HAS_RAW>
__global__ void __launch_bounds__(128)
gemm_wmma_kernel(const float* __restrict__ A, int lda, long aOuter, long aInner,
                 const float* __restrict__ B, int ldb, long bOuter, long bInner,
                 float* __restrict__ O, int ldo, long oOuter, long oInner,
                 float* __restrict__ Oraw,
                 const float* __restrict__ bias, const float* __restrict__ addvec,
                 int K) {
  __shared__ float As[64 * 20];
  __shared__ float Bs[64 * 20];

  int lane = threadIdx.x;            // 0..31 (wave32)
  int wave = threadIdx.y;            // 0..3
  int t    = wave * 32 + lane;       // 0..127
  int zo   = blockIdx.z >> 2;       // z / C
  int zi   = blockIdx.z & 3;        // z % C
  const float* Ab = A + (long)zo * aOuter + (long)zi * aInner;
  const float* Bb = B + (long)zo * bOuter + (long)zi * bInner;

  int mblk = blockIdx.y * 64;
  int nblk = blockIdx.x * 64;
  int mw   = (wave >> 1) * 32;       // wave sub-tile origin within block tile
  int nw   = (wave & 1) * 32;

  v8f c00 = {}, c01 = {}, c10 = {}, c11 = {};

  for (int kc = 0; kc < K; kc += 16) {
    __syncthreads();
    // Cooperative staging: A = 64 rows x 16 k (256 float4), B = 16 k x 64 n (256 float4).
#pragma unroll
    for (int rep = 0; rep < 2; ++rep) {
      int idx = t + rep * 128;
      // A: row-major copy
      int ar = idx >> 2, ac = (idx & 3) << 2;
      float4 av = *(const float4*)(Ab + (long)(mblk + ar) * lda + kc + ac);
      *(float4*)(&As[ar * 20 + ac]) = av;
      // B: transpose into Bs[n][k]
      int br = idx >> 4, bc = (idx & 15) << 2;
      float4 bv = *(const float4*)(Bb + (long)(kc + br) * ldb + nblk + bc);
      Bs[(bc + 0) * 20 + br] = bv.x;
      Bs[(bc + 1) * 20 + br] = bv.y;
      Bs[(bc + 2) * 20 + br] = bv.z;
      Bs[(bc + 3) * 20 + br] = bv.w;
    }
    __syncthreads();

#pragma unroll
    for (int kl = 0; kl < 16; kl += 4) {
      // f32 16x16x4 fragment layout: lanes 0-15 hold K=0,1; lanes 16-31 hold K=2,3.
      int kf = kl + ((lane >> 4) << 1);
      int rr = lane & 15;
      v2f a0 = *(const v2f*)(&As[(mw + rr) * 20 + kf]);
      v2f a1 = *(const v2f*)(&As[(mw + 16 + rr) * 20 + kf]);
      v2f b0 = *(const v2f*)(&Bs[(nw + rr) * 20 + kf]);
      v2f b1 = *(const v2f*)(&Bs[(nw + 16 + rr) * 20 + kf]);
      c00 = __builtin_amdgcn_wmma_f32_16x16x4_f32(false, a0, false, b0, (short)0, c00, false, false);
      c01 = __builtin_amdgcn_wmma_f32_16x16x4_f32(false, a0, false, b1, (short)0, c01, false, false);
      c10 = __builtin_amdgcn_wmma_f32_16x16x4_f32(false, a1, false, b0, (short)0, c10, false, false);
      c11 = __builtin_amdgcn_wmma_f32_16x16x4_f32(false, a1, false, b1, (short)0, c11, false, false);
    }
  }

  float* Ob = O + (long)zo * oOuter + (long)zi * oInner;
  float* Rb = HAS_RAW ? (Oraw + (long)zo * oOuter + (long)zi * oInner) : nullptr;
  int m0 = mblk + mw, n0 = nblk + nw;
  store_tile<MODE, HAS_RAW>(c00, Ob, ldo, Rb, bias, addvec, m0,      n0,      lane);
  store_tile<MODE, HAS_RAW>(c01, Ob, ldo, Rb, bias, addvec, m0,      n0 + 16, lane);
  store_tile<MODE, HAS_RAW>(c10, Ob, ldo, Rb, bias, addvec, m0 + 16, n0,      lane);
  store_tile<MODE, HAS_RAW>(c11, Ob, ldo, Rb, bias, addvec, m0 + 16, n0 + 16, lane);
}

// ---------------- Final fused broadcast + tanh + reduce ----------------
// out[b,p,c,a] = sum_k tanh(pref[b,p,c,k] + targ[b,a,c,k]) * w_out[k]
// This is the compute bottleneck (B*S*S*C*H = 151M transcendental lane-ops);
// pref row + w_out staged in LDS, each thread owns one 'a' and streams its
// L2-resident targ row with 128-bit loads. Inner loop: add + v_exp + v_rcp + fma.
__global__ void __launch_bounds__(128)
final_reduce_kernel(const float* __restrict__ pref, const float* __restrict__ targ,
                    const float* __restrict__ wout, float* __restrict__ out) {
  __shared__ float sp[HH];
  __shared__ float sw[HH];
  int tid = threadIdx.x;                 // 0..127
  int p   = blockIdx.y;                  // 0..S-1
  int bc  = blockIdx.z;                  // 0..B*C-1
  int b   = bc >> 2;
  int c   = bc & 3;

  long prow = ((long)(b * SS + p) * CC + c) * HH;
  sp[tid] = pref[prow + tid];
  sw[tid] = wout[tid];
  __syncthreads();

  int a = blockIdx.x * 128 + tid;        // 3*128 == S, no bounds check needed
  const float* ta = targ + ((long)(b * SS + a) * CC + c) * HH;
  __builtin_prefetch(ta, 0, 0);          // global_prefetch_b8

  float acc = 0.0f;
#pragma unroll 8
  for (int k = 0; k < HH; k += 4) {
    float4 tv = *(const float4*)(ta + k);
    acc += tanhf_fast(sp[k + 0] + tv.x) * sw[k + 0];
    acc += tanhf_fast(sp[k + 1] + tv.y) * sw[k + 1];
    acc += tanhf_fast(sp[k + 2] + tv.z) * sw[k + 2];
    acc += tanhf_fast(sp[k + 3] + tv.w) * sw[k + 3];
  }
  out[((long)(b * SS + p) * CC + c) * SS + a] = acc;
}

// ---------------- Host launch ----------------
extern "C" void kernel_launch(void* const* d_in, const int* in_sizes, int n_in,
                              void* d_out, int out_size, void* d_ws, size_t ws_size,
                              hipStream_t stream) {
  const float* seq  = (const float*)d_in[0]; // (B,S,D)
  const float* base = (const float*)d_in[1]; // (B,S,C,S)
  const float* Wp   = (const float*)d_in[2]; // (D, C*H)
  const float* bp   = (const float*)d_in[3]; // (C*H,)
  const float* Ua   = (const float*)d_in[4]; // (D, C*H)
  const float* ba   = (const float*)d_in[5]; // (C*H,)
  const float* Wmid = (const float*)d_in[6]; // (H*(C+2), H)
  const float* bmid = (const float*)d_in[7]; // (H,)
  const float* wout = (const float*)d_in[8]; // (H,)
  float* out = (float*)d_out;

  // Workspace layout (floats). Total = 2,457,600 floats = 9.83 MB (L2-resident).
  float* ws      = (float*)d_ws;
  float* tanh_hp = ws;                     // B*S*C*H = 393216
  float* ha_raw  = ws + 393216;            // 393216
  float* tanh_ha = ws + 786432;            // 393216
  float* tanhref = ws + 1179648;           // 393216
  float* targ    = ws + 1572864;           // 393216
  float* tref    = ws + 1966080;           // B*S*H = 98304
  float* pref    = ws + 2064384;           // 393216

  dim3 blk(32, 4);

  // h_p = seq@W_prd + b ; store tanh(h_p).   M=768 K=768 N=512
  gemm_wmma_kernel<MODE_TANH | MODE_BIAS, false><<<dim3(8, 12, 1), blk, 0, stream>>>(
      seq, DD, 0, 0, Wp, CC * HH, 0, 0, tanh_hp, CC * HH, 0, 0,
      nullptr, bp, nullptr, DD);

  // h_a = seq@U_arg + b ; store raw h_a AND tanh(h_a).
  gemm_wmma_kernel<MODE_TANH | MODE_BIAS, true><<<dim3(8, 12, 1), blk, 0, stream>>>(
      seq, DD, 0, 0, Ua, CC * HH, 0, 0, tanh_ha, CC * HH, 0, 0,
      ha_raw, ba, nullptr, DD);

  // ref[b,:,c,:] = base_score[b,:,c,:] @ h_a[b,:,c,:] ; store tanh(ref).
  // 8 batches (b,c): M=384 K=384 N=128
  gemm_wmma_kernel<MODE_TANH, false><<<dim3(2, 6, BB * CC), blk, 0, stream>>>(
      base, CC * SS, (long)SS * CC * SS, SS,
      ha_raw, CC * HH, (long)SS * CC * HH, HH,
      tanhref, CC * HH, (long)SS * CC * HH, HH,
      nullptr, nullptr, nullptr, SS);

  // t_ref = tanh(ref) @ W3.   M=768 K=512 N=128   (W3 = W_mid rows 2H..6H)
  gemm_wmma_kernel<0, false><<<dim3(2, 12, 1), blk, 0, stream>>>(
      tanhref, CC * HH, 0, 0, Wmid + 2 * HH * HH, HH, 0, 0, tref, HH, 0, 0,
      nullptr, nullptr, nullptr, CC * HH);

  // pref = tanh(h_p)@W1 + t_ref(row/C) + b_mid.   M=3072 K=128 N=128
  gemm_wmma_kernel<MODE_BIAS | MODE_ADDV, false><<<dim3(2, 48, 1), blk, 0, stream>>>(
      tanh_hp, HH, 0, 0, Wmid, HH, 0, 0, pref, HH, 0, 0,
      nullptr, bmid, tref, HH);

  // targ = tanh(h_a) @ W2.   M=3072 K=128 N=128
  gemm_wmma_kernel<0, false><<<dim3(2, 48, 1), blk, 0, stream>>>(
      tanh_ha, HH, 0, 0, Wmid + HH * HH, HH, 0, 0, targ, HH, 0, 0,
      nullptr, nullptr, nullptr, HH);

  // out[b,p,c,a] = sum_k tanh(pref + targ) * w_out
  final_reduce_kernel<<<dim3(SS / 128, SS, BB * CC), dim3(128), 0, stream>>>(
      pref, targ, wout, out);
}